// SparseCNN_28656021799500
// MI455X (gfx1250) — compile-verified
//
#include <hip/hip_runtime.h>
#include <hip/hip_bf16.h>

typedef __attribute__((ext_vector_type(2))) float v2f;
typedef __attribute__((ext_vector_type(8))) float v8f;

#if defined(__AMDGCN__) && __has_builtin(__builtin_amdgcn_global_load_async_to_lds_b32)
#define HAVE_ASYNC_LDS 1
#else
#define HAVE_ASYNC_LDS 0
#endif

#if HAVE_ASYNC_LDS
typedef __attribute__((address_space(1))) int g_as_int;   // global
typedef __attribute__((address_space(3))) int l_as_int;   // LDS
#endif

__device__ __forceinline__ void wait_async_zero() {
#if HAVE_ASYNC_LDS
#if __has_builtin(__builtin_amdgcn_s_wait_asynccnt)
    __builtin_amdgcn_s_wait_asynccnt(0);
#else
    asm volatile("s_wait_asynccnt 0x0" ::: "memory");
#endif
#endif
}

// ---------------------------------------------------------------------------
// Sparse conv via gather + V_WMMA_F32_16X16X4_F32 (fp32 in/out, matches ref).
// Block = 256 threads = 8 wave32s; each wave computes a 16-row x CO tile.
// Weights for tap k are staged into LDS in a swizzled layout
//   W[k][kr][n] -> lds[ ((kr>>1)*CO + n)*2 + (kr&1) ]
// so a lane's B-fragment (two consecutive K-rows) is a single ds_load_b64.
// Staging is double-buffered: tap k+1 is prefetched via
// GLOBAL_LOAD_ASYNC_TO_LDS_B32 (ASYNCcnt) while the WMMAs for tap k run.
//
// Fragment layouts (ISA 7.12.2, wave32):
//  A 16x4 f32: lanes0-15 -> M=lane, K={0,1}; lanes16-31 -> M=lane-16, K={2,3}
//  B 4x16 f32: lanes0-15 -> N=lane, K={0,1}; lanes16-31 -> N=lane-16, K={2,3}
//  C/D 16x16:  VGPR j -> M=j (lanes0-15) / M=j+8 (lanes16-31), N=lane&15
// ---------------------------------------------------------------------------
template<int CI, int CO, int KK>
__global__ __launch_bounds__(256) void sconv_wmma(
    const float* __restrict__ feat, const int* __restrict__ rb,
    const float* __restrict__ W, float* __restrict__ out, int Nout)
{
    __shared__ float wbuf[2][CI * CO];
    const int tid  = threadIdx.x;
    const int lane = tid & 31;
    const int wave = tid >> 5;
    const int tileBase = (blockIdx.x * 8 + wave) * 16;
    const int m  = lane & 15;
    const bool hi = lane >= 16;

    auto stage = [&](int buf, int kk) {
        const float* Wk = W + (size_t)kk * (CI * CO);
        for (int i = tid; i < CI * CO; i += 256) {
            const int kr = i / CO;
            const int n  = i - kr * CO;
            const int d  = ((kr >> 1) * CO + n) * 2 + (kr & 1);
#if HAVE_ASYNC_LDS
            __builtin_amdgcn_global_load_async_to_lds_b32(
                (g_as_int*)(Wk + i),            // addrspacecast generic->AS1
                (l_as_int*)&wbuf[buf][d],       // addrspacecast generic->AS3
                0, 0);
#else
            wbuf[buf][d] = Wk[i];
#endif
        }
    };

    v8f acc[CO / 16];
#pragma unroll
    for (int t = 0; t < CO / 16; ++t)
        acc[t] = (v8f){0.f, 0.f, 0.f, 0.f, 0.f, 0.f, 0.f, 0.f};

    stage(0, 0);   // prefetch tap 0

    for (int kk = 0; kk < KK; ++kk) {
        const int cur = kk & 1;
        wait_async_zero();        // weights for tap kk landed in wbuf[cur]
        __syncthreads();          // all waves done reading wbuf[cur^1]
        if (kk + 1 < KK)
            stage(cur ^ 1, kk + 1);   // overlap next staging with compute

        if (tileBase < Nout) {   // wave-uniform: EXEC stays all-1s for WMMA
            const int row = tileBase + m;
            const int idx = (row < Nout) ? rb[kk * Nout + row] : -1;
            const float* src = feat + (long)(idx < 0 ? 0 : idx) * CI;
#pragma unroll
            for (int kc = 0; kc < CI / 4; ++kc) {
                float ax = 0.f, ay = 0.f;
                if (idx >= 0) {
                    const float* p = src + kc * 4 + (hi ? 2 : 0);
                    ax = p[0];
                    ay = p[1];
                }
                v2f a; a.x = ax; a.y = ay;
                const int pr = kc * 2 + (hi ? 1 : 0);   // K-row pair index
#pragma unroll
                for (int t = 0; t < CO / 16; ++t) {
                    const v2f b =
                        *(const v2f*)&wbuf[cur][2 * (pr * CO + t * 16 + m)];
                    acc[t] = __builtin_amdgcn_wmma_f32_16x16x4_f32(
                        false, a, false, b, (short)0, acc[t], false, false);
                }
            }
        }
    }

    if (tileBase < Nout) {
#pragma unroll
        for (int t = 0; t < CO / 16; ++t) {
#pragma unroll
            for (int j = 0; j < 8; ++j) {
                const int row = tileBase + j + (hi ? 8 : 0);
                if (row < Nout)
                    out[(long)row * CO + t * 16 + m] = acc[t][j];
            }
        }
    }
}

// ---------------------------------------------------------------------------
// BatchNorm (training-mode, biased var, eps=1e-3) reduction: one block/channel.
// Emits scale = g/sqrt(var+eps), shift = b - mean*scale.
// ---------------------------------------------------------------------------
__global__ __launch_bounds__(256) void bn_reduce(
    const float* __restrict__ x, const float* __restrict__ g,
    const float* __restrict__ b, float* __restrict__ scale,
    float* __restrict__ shift, int N, int co)
{
    __shared__ float ssum[256];
    __shared__ float ssq[256];
    const int c = blockIdx.x;
    float s = 0.f, q = 0.f;
    for (int n = threadIdx.x; n < N; n += 256) {
        const float v = x[(long)n * co + c];
        s += v;
        q += v * v;
    }
    ssum[threadIdx.x] = s;
    ssq[threadIdx.x]  = q;
    __syncthreads();
    for (int off = 128; off > 0; off >>= 1) {
        if (threadIdx.x < off) {
            ssum[threadIdx.x] += ssum[threadIdx.x + off];
            ssq[threadIdx.x]  += ssq[threadIdx.x + off];
        }
        __syncthreads();
    }
    if (threadIdx.x == 0) {
        const float invN = 1.0f / (float)N;
        const float mean = ssum[0] * invN;
        float var = ssq[0] * invN - mean * mean;
        var = var < 0.f ? 0.f : var;
        const float sc = g[c] * rsqrtf(var + 1e-3f);
        scale[c] = sc;
        shift[c] = b[c] - mean * sc;
    }
}

__global__ void bn_apply(const float* __restrict__ x,
                         const float* __restrict__ scale,
                         const float* __restrict__ shift,
                         float* __restrict__ y, int total, int co)
{
    const int i = blockIdx.x * blockDim.x + threadIdx.x;
    if (i < total) {
        const int c = i & (co - 1);   // co is a power of two (16/32/64)
        y[i] = fmaxf(fmaf(x[i], scale[c], shift[c]), 0.0f);
    }
}

// coords (b,z,y,x) int32 -> xyz = (x*0.05*s, y*0.05*s - 40, z*0.1*s - 3)
__global__ void xyz_kernel(const int* __restrict__ c, float* __restrict__ o,
                           int N, float s)
{
    const int n = blockIdx.x * blockDim.x + threadIdx.x;
    if (n < N) {
        o[(long)n * 3 + 0] = (float)c[(long)n * 4 + 3] * (0.05f * s);
        o[(long)n * 3 + 1] = (float)c[(long)n * 4 + 2] * (0.05f * s) - 40.0f;
        o[(long)n * 3 + 2] = (float)c[(long)n * 4 + 1] * (0.10f * s) - 3.0f;
    }
}

extern "C" void kernel_launch(void* const* d_in, const int* in_sizes, int n_in,
                              void* d_out, int out_size, void* d_ws, size_t ws_size,
                              hipStream_t stream)
{
    (void)n_in; (void)out_size; (void)ws_size;

    const float* feat = (const float*)d_in[0];
    auto Wp = [&](int l) { return (const float*)d_in[1 + 3 * l]; };
    auto Gp = [&](int l) { return (const float*)d_in[2 + 3 * l]; };
    auto Bp = [&](int l) { return (const float*)d_in[3 + 3 * l]; };

    const int* coords0 = (const int*)d_in[43];
    const int* coords1 = (const int*)d_in[44];
    const int* coords2 = (const int*)d_in[45];
    const int* coords3 = (const int*)d_in[46];
    const int* rb0  = (const int*)d_in[47];
    const int* rbc1 = (const int*)d_in[48];
    const int* rb1  = (const int*)d_in[49];
    const int* rbc2 = (const int*)d_in[50];
    const int* rb2  = (const int*)d_in[51];
    const int* rbc3 = (const int*)d_in[52];
    const int* rb3  = (const int*)d_in[53];
    const int* rbc4 = (const int*)d_in[54];

    const int N0 = in_sizes[43] / 4;
    const int N1 = in_sizes[44] / 4;
    const int N2 = in_sizes[45] / 4;
    const int N3 = in_sizes[46] / 4;
    const int N4 = in_sizes[54] / 3;

    // ---- output layout (flat fp32 concat in return order) ----
    float* out = (float*)d_out;
    float* o_xyz0 = out;
    float* o_f0   = o_xyz0 + (size_t)3 * N0;
    float* o_xyz1 = o_f0   + (size_t)4 * N0;
    float* o_f1   = o_xyz1 + (size_t)3 * N1;
    float* o_xyz2 = o_f1   + (size_t)32 * N1;
    float* o_f2   = o_xyz2 + (size_t)3 * N2;
    float* o_xyz3 = o_f2   + (size_t)64 * N2;
    float* o_f3   = o_xyz3 + (size_t)3 * N3;
    float* o_f4   = o_f3   + (size_t)64 * N3;

    // ---- workspace layout ----
    int maxN = N0;
    if (N1 > maxN) maxN = N1;
    if (N2 > maxN) maxN = N2;
    if (N3 > maxN) maxN = N3;
    if (N4 > maxN) maxN = N4;
    const size_t cap = (size_t)maxN * 64;
    float* raw   = (float*)d_ws;
    float* hA    = raw + cap;
    float* hB    = hA + cap;
    float* scale = hB + cap;
    float* shift = scale + 64;

    // ---- coordinate outputs + f0 passthrough ----
    xyz_kernel<<<(N0 + 255) / 256, 256, 0, stream>>>(coords0, o_xyz0, N0, 1.0f);
    xyz_kernel<<<(N1 + 255) / 256, 256, 0, stream>>>(coords1, o_xyz1, N1, 2.0f);
    xyz_kernel<<<(N2 + 255) / 256, 256, 0, stream>>>(coords2, o_xyz2, N2, 4.0f);
    xyz_kernel<<<(N3 + 255) / 256, 256, 0, stream>>>(coords3, o_xyz3, N3, 8.0f);
    (void)hipMemcpyAsync(o_f0, feat, (size_t)4 * N0 * sizeof(float),
                         hipMemcpyDeviceToDevice, stream);

    auto bn = [&](int l, int Nout, int co, float* dest) {
        bn_reduce<<<co, 256, 0, stream>>>(raw, Gp(l), Bp(l), scale, shift, Nout, co);
        const int tot = Nout * co;
        bn_apply<<<(tot + 255) / 256, 256, 0, stream>>>(raw, scale, shift, dest, tot, co);
    };
    auto blocks = [](int Nout) { return (Nout + 127) / 128; };

    // ---- layer chain ----
    // stage 1
    sconv_wmma<4, 16, 27><<<blocks(N0), 256, 0, stream>>>(feat, rb0, Wp(0), raw, N0);
    bn(0, N0, 16, hA);
    sconv_wmma<16, 16, 27><<<blocks(N0), 256, 0, stream>>>(hA, rb0, Wp(1), raw, N0);
    bn(1, N0, 16, hB);
    sconv_wmma<16, 32, 27><<<blocks(N1), 256, 0, stream>>>(hB, rbc1, Wp(2), raw, N1);
    bn(2, N1, 32, o_f1);
    // stage 2
    sconv_wmma<32, 32, 27><<<blocks(N1), 256, 0, stream>>>(o_f1, rb1, Wp(3), raw, N1);
    bn(3, N1, 32, hA);
    sconv_wmma<32, 32, 27><<<blocks(N1), 256, 0, stream>>>(hA, rb1, Wp(4), raw, N1);
    bn(4, N1, 32, hB);
    sconv_wmma<32, 64, 27><<<blocks(N2), 256, 0, stream>>>(hB, rbc2, Wp(5), raw, N2);
    bn(5, N2, 64, o_f2);
    // stage 3
    sconv_wmma<64, 64, 27><<<blocks(N2), 256, 0, stream>>>(o_f2, rb2, Wp(6), raw, N2);
    bn(6, N2, 64, hA);
    sconv_wmma<64, 64, 27><<<blocks(N2), 256, 0, stream>>>(hA, rb2, Wp(7), raw, N2);
    bn(7, N2, 64, hB);
    sconv_wmma<64, 64, 27><<<blocks(N2), 256, 0, stream>>>(hB, rb2, Wp(8), raw, N2);
    bn(8, N2, 64, hA);
    sconv_wmma<64, 64, 27><<<blocks(N3), 256, 0, stream>>>(hA, rbc3, Wp(9), raw, N3);
    bn(9, N3, 64, o_f3);
    // stage 4
    sconv_wmma<64, 64, 27><<<blocks(N3), 256, 0, stream>>>(o_f3, rb3, Wp(10), raw, N3);
    bn(10, N3, 64, hA);
    sconv_wmma<64, 64, 27><<<blocks(N3), 256, 0, stream>>>(hA, rb3, Wp(11), raw, N3);
    bn(11, N3, 64, hB);
    sconv_wmma<64, 64, 27><<<blocks(N3), 256, 0, stream>>>(hB, rb3, Wp(12), raw, N3);
    bn(12, N3, 64, hA);
    sconv_wmma<64, 64, 3><<<blocks(N4), 256, 0, stream>>>(hA, rbc4, Wp(13), raw, N4);
    bn(13, N4, 64, o_f4);
}